// RALoss_44487271252553
// MI455X (gfx1250) — compile-verified
//
#include <hip/hip_runtime.h>
#include <math.h>

// ---- problem geometry (SHAPE = [4, 64, 512, 512], f32) ----
#define L_DIM     4
#define B_DIM     64
#define MAPS      (L_DIM * B_DIM)     // 256 independent (L,B) reductions
#define MAP_ELEMS (512 * 512)         // 262144 f32 per map
#define MAP_VEC4  (MAP_ELEMS / 4)     // 65536 float4 per map

#define TPB       256                 // 8 wave32 per block
#define CHUNKS    32                  // blocks per map
#define VEC_ITERS 8                   // float4 per thread: 32*256*8 = 65536  ✓

typedef float v2f __attribute__((ext_vector_type(2)));
typedef float v4f __attribute__((ext_vector_type(4)));
typedef float v8f __attribute__((ext_vector_type(8)));

// -----------------------------------------------------------------------------
// Stage 1: each block reduces a 64 KB slice of one (L,B) map.
//   - b128 non-temporal streaming loads (data is touched exactly once; keep
//     the 192 MB L2 clean)
//   - per-lane f32 accumulation, then one V_WMMA_F32_16X16X4_F32 with an
//     all-ones B matrix collapses the 32 lane partials: D[m][n] =
//     partial[m] + partial[m+16]  (A layout: VGPR0 = K0 on lanes 0-15,
//     K2 on lanes 16-31; VGPR1 forced to 0)
//   - deterministic: per-wave slot in LDS, thread 0 writes the block partial
// -----------------------------------------------------------------------------
__global__ __launch_bounds__(TPB) void ra_partial_kernel(
    const float* __restrict__ amax,
    const float* __restrict__ aorg,
    float* __restrict__ partials /* [MAPS][CHUNKS] */) {

  const int map   = blockIdx.y;   // 0..255
  const int chunk = blockIdx.x;   // 0..31
  const int tid   = threadIdx.x;

  const v4f* __restrict__ A = (const v4f*)amax;
  const v4f* __restrict__ B = (const v4f*)aorg;

  const size_t base4 =
      (size_t)map * MAP_VEC4 + (size_t)chunk * (TPB * VEC_ITERS) + tid;

  float lane_sum = 0.0f;
#pragma unroll
  for (int it = 0; it < VEC_ITERS; ++it) {
    const size_t i = base4 + (size_t)it * TPB;          // coalesced b128
    v4f a = __builtin_nontemporal_load(&A[i]);          // TH = NT streaming
    v4f b = __builtin_nontemporal_load(&B[i]);
    lane_sum += (a.x - b.x) + (a.y - b.y) + (a.z - b.z) + (a.w - b.w);
  }

  // ---- wave32 cross-lane reduce via WMMA (ones-matrix trick) ----
  // EXEC is all ones here: full blocks, no divergence above.
  v2f a2;   a2.x = lane_sum; a2.y = 0.0f;   // A[m][0]=p[m], A[m][2]=p[m+16]
  v2f ones; ones.x = 1.0f;  ones.y = 1.0f;  // B[k][n] = 1
  v8f c = {};
  c = __builtin_amdgcn_wmma_f32_16x16x4_f32(
      /*neg_a=*/false, a2, /*neg_b=*/false, ones,
      /*c_mod=*/(short)0, c, /*reuse_a=*/false, /*reuse_b=*/false);

  // C/D layout: lane n (0-15) VGPR v = D[v][n]   (rows 0..7)
  //             lane n (16-31) VGPR v = D[v+8][n-16] (rows 8..15)
  // Each D[m][n] = partial[m] + partial[m+16]; columns are replicated, so:
  float half = c[0] + c[1] + c[2] + c[3] + c[4] + c[5] + c[6] + c[7];
  // lane 0 holds rows 0..7, lane 16 holds rows 8..15 -> add them.
  float wave_total = half + __shfl_down(half, 16, 32);

  __shared__ float wsum[TPB / 32];
  const int lane = tid & 31;
  const int wid  = tid >> 5;
  if (lane == 0) wsum[wid] = wave_total;
  __syncthreads();

  if (tid == 0) {
    float s = 0.0f;
#pragma unroll
    for (int w = 0; w < TPB / 32; ++w) s += wsum[w];   // deterministic order
    partials[(size_t)map * CHUNKS + chunk] = s;        // overwrite: no init needed
  }
}

// -----------------------------------------------------------------------------
// Stage 2: one block, 256 threads (one per map). diff = Σ chunk partials
// (sequential, deterministic), contribution = 10*exp(-diff)/B_DIM, then a
// single-thread ordered sum -> d_out[0]. Cost is ~8K flops: irrelevant.
// -----------------------------------------------------------------------------
__global__ __launch_bounds__(MAPS) void ra_finalize_kernel(
    const float* __restrict__ partials, float* __restrict__ out) {

  __shared__ float vals[MAPS];
  const int t = threadIdx.x;   // map index = i*B_DIM + j

  float diff = 0.0f;
#pragma unroll
  for (int cidx = 0; cidx < CHUNKS; ++cidx)
    diff += partials[(size_t)t * CHUNKS + cidx];

  vals[t] = (10.0f / (float)B_DIM) * expf(-diff);
  __syncthreads();

  if (t == 0) {
    float s = 0.0f;
    for (int i = 0; i < MAPS; ++i) s += vals[i];       // deterministic order
    out[0] = s;
  }
}

// -----------------------------------------------------------------------------
extern "C" void kernel_launch(void* const* d_in, const int* in_sizes, int n_in,
                              void* d_out, int out_size, void* d_ws, size_t ws_size,
                              hipStream_t stream) {
  (void)in_sizes; (void)n_in; (void)out_size; (void)ws_size;

  const float* amax = (const float*)d_in[0];
  const float* aorg = (const float*)d_in[1];
  float* partials   = (float*)d_ws;          // 256*32 floats = 32 KB scratch
  float* out        = (float*)d_out;

  dim3 grid(CHUNKS, MAPS);                   // 8192 blocks x 8 wave32 = deep MLP
  ra_partial_kernel<<<grid, TPB, 0, stream>>>(amax, aorg, partials);
  ra_finalize_kernel<<<1, MAPS, 0, stream>>>(partials, out);
}